// HGNN_32134945309312
// MI455X (gfx1250) — compile-verified
//
#include <hip/hip_runtime.h>
#include <hip/hip_bf16.h>

// ---------------------------------------------------------------------------
// Frontier-sparse GCN mutation-effect predictor for MI455X (gfx1250).
// Output depends only on h2[argmax(mask)] -> compute the 2-hop in-frontier
// only. Edge-list streaming passes dominate (~50MB @ 23.3TB/s, pass 2 mostly
// L2-resident since 32MB < 192MB L2); the small dense GEMMs use
// V_WMMA_F32_16X16X4_F32 (fp32 WMMA path, wave32).
// ---------------------------------------------------------------------------

typedef float v2f __attribute__((ext_vector_type(2)));
typedef float v8f __attribute__((ext_vector_type(8)));

#define IN_DIM 128
#define HID    256
#define CAP0   2048     // max |S0| rows (expected ~290)
#define CAP1   256      // max |S1| rows (expected ~17)
#define CAPE1  1024     // max edges into m (expected ~16)
#define CAPE2  16384    // max edges into S1 (expected ~280)

// scal[] layout: 0=m, 1=cnt1, 2=cnt2, 3=n1, 4=n0, 5=maskval(float bits)

__device__ __forceinline__ int imin(int a, int b) { return a < b ? a : b; }

// ---- init all per-call mutable state -------------------------------------
__global__ void k_init(float* deg, int* flag1, int* flag0, int* scal, int n) {
    int i = blockIdx.x * blockDim.x + threadIdx.x;
    if (i < n) { deg[i] = 1.0f; flag1[i] = 0; flag0[i] = 0; }  // deg=1: self loop
    if (i < 16) scal[i] = 0;
}

// ---- find one-hot index m ------------------------------------------------
__global__ void k_findm(const float* __restrict__ mask, int* flag1, int* scal, int n) {
    int i = blockIdx.x * blockDim.x + threadIdx.x;
    if (i < n && mask[i] > 0.5f) {
        scal[0] = i;
        ((float*)scal)[5] = mask[i];
        flag1[i] = 1;
    }
}

// ---- pass 1 over edges: degrees + collect edges into m, mark S1 ----------
// Row index is loaded ONLY on the rare col==m hit: 12 B/edge streamed, not 20.
__global__ void k_edge1(const long long* __restrict__ ei, const float* __restrict__ ew,
                        float* deg, int* flag1, int* scal,
                        int* rowL1, float* ewL1, int E) {
    int e = blockIdx.x * blockDim.x + threadIdx.x;
    if (e >= E) return;
    int c = (int)ei[(long long)E + e];
    float w = ew[e];
    atomicAdd(&deg[c], w);
    if (c == scal[0]) {
        int r = (int)ei[e];                    // cold path: ~16 of 1.6M edges
        int idx = atomicAdd(&scal[1], 1);
        if (idx < CAPE1) { rowL1[idx] = r; ewL1[idx] = w; }
        flag1[r] = 1;
    }
}

// ---- compact S1 -> list/map, seed S0 flags -------------------------------
__global__ void k_compact1(const int* __restrict__ flag1, int* flag0, int* map1,
                           int* list1, int* scal, int n) {
    int i = blockIdx.x * blockDim.x + threadIdx.x;
    if (i >= n) return;
    if (flag1[i]) {
        int s = atomicAdd(&scal[3], 1);
        if (s < CAP1) { list1[s] = i; map1[i] = s; }
        flag0[i] = 1;
    }
}

// ---- pass 2 over edges: collect edges into S1, mark S0 -------------------
__global__ void k_edge2(const long long* __restrict__ ei, const float* __restrict__ ew,
                        const int* __restrict__ flag1, int* flag0, int* scal,
                        int* rowL2, int* colL2, float* ewL2, int E) {
    int e = blockIdx.x * blockDim.x + threadIdx.x;
    if (e >= E) return;
    int c = (int)ei[(long long)E + e];
    if (flag1[c]) {
        int r = (int)ei[e];                    // cold path: ~280 of 1.6M edges
        int idx = atomicAdd(&scal[2], 1);
        if (idx < CAPE2) { rowL2[idx] = r; colL2[idx] = c; ewL2[idx] = ew[e]; }
        flag0[r] = 1;
    }
}

// ---- compact S0 ----------------------------------------------------------
__global__ void k_compact0(const int* __restrict__ flag0, int* map0, int* list0,
                           int* scal, int n) {
    int i = blockIdx.x * blockDim.x + threadIdx.x;
    if (i >= n) return;
    if (flag0[i]) {
        int s = atomicAdd(&scal[4], 1);
        if (s < CAP0) { list0[s] = i; map0[i] = s; }
    }
}

// ---- gather x rows for S0 (zero padding) ---------------------------------
__global__ void k_gatherx(const float* __restrict__ x, const int* __restrict__ list0,
                          const int* __restrict__ scal, float* __restrict__ Xg) {
    int slot = blockIdx.x, t = threadIdx.x;
    int n0 = imin(scal[4], CAP0);
    float v = 0.0f;
    if (slot < n0) v = x[(long long)list0[slot] * IN_DIM + t];
    Xg[(long long)slot * IN_DIM + t] = v;
}

// ---- WMMA fp32 GEMM: C[rows x 256] = A[rows x K] * B[K x 256] ------------
// One wave per 16x16 output tile. V_WMMA_F32_16X16X4_F32, K stepped by 4.
// A frag (16x4, 2 VGPRs): lanes 0-15 -> K={k,k+1}, lanes 16-31 -> K={k+2,k+3}
// C frag (16x16, 8 VGPRs): vgpr i -> M=i (lanes 0-15), M=i+8 (lanes 16-31)
__global__ void __launch_bounds__(32)
k_wmma_gemm(const float* __restrict__ A, const float* __restrict__ B,
            float* __restrict__ C, int K, const int* __restrict__ nvalid, int maxrows) {
    int nv = imin(*nvalid, maxrows);
    if ((int)blockIdx.x * 16 >= nv) return;   // uniform branch: EXEC stays all-1s
    const int lane = threadIdx.x;
    const int half = lane >> 4;               // 0 or 1
    const int l15  = lane & 15;
    const long long arow = (long long)(blockIdx.x * 16 + l15) * K;
    const int ncol = blockIdx.y * 16 + l15;
    v8f acc = {};
    for (int k = 0; k < K; k += 4) {
        int ka = k + 2 * half;
        v2f a, b;
        a.x = A[arow + ka];
        a.y = A[arow + ka + 1];
        b.x = B[(long long)ka * HID + ncol];
        b.y = B[(long long)(ka + 1) * HID + ncol];
        acc = __builtin_amdgcn_wmma_f32_16x16x4_f32(
            false, a, false, b, (short)0, acc, false, false);
    }
#pragma unroll
    for (int i = 0; i < 8; ++i) {
        int row = blockIdx.x * 16 + i + 8 * half;
        C[(long long)row * HID + ncol] = acc[i];
    }
}

// ---- layer1 aggregation: init with bias + self-loop term -----------------
__global__ void k_agg1init(const float* __restrict__ P, const float* __restrict__ deg,
                           const float* __restrict__ b1, const int* __restrict__ list1,
                           const int* __restrict__ map0, const int* __restrict__ scal,
                           float* __restrict__ h1) {
    int slot = blockIdx.x, c = threadIdx.x;
    int n1 = imin(scal[3], CAP1);
    float v = 0.0f;
    if (slot < n1) {
        int u = list1[slot];
        float dv = rsqrtf(deg[u]);
        v = b1[c] + dv * dv * P[(long long)map0[u] * HID + c];
    }
    h1[(long long)slot * HID + c] = v;
}

// ---- layer1 aggregation: scatter edge contributions ----------------------
__global__ void k_agg1edge(const float* __restrict__ P, const float* __restrict__ deg,
                           const int* __restrict__ rowL2, const int* __restrict__ colL2,
                           const float* __restrict__ ewL2, const int* __restrict__ map0,
                           const int* __restrict__ map1, const int* __restrict__ scal,
                           float* __restrict__ h1) {
    int e = blockIdx.x;
    int cnt = imin(scal[2], CAPE2);
    if (e >= cnt) return;
    int r = rowL2[e], c = colL2[e];
    float nrm = rsqrtf(deg[r]) * ewL2[e] * rsqrtf(deg[c]);
    atomicAdd(&h1[(long long)map1[c] * HID + threadIdx.x],
              nrm * P[(long long)map0[r] * HID + threadIdx.x]);
}

__global__ void k_relu(float* h, int n) {
    int i = blockIdx.x * blockDim.x + threadIdx.x;
    if (i < n) h[i] = fmaxf(h[i], 0.0f);
}

// ---- layer2 aggregation at node m only -> z[256] -------------------------
__global__ void k_final(const float* __restrict__ Q, const float* __restrict__ deg,
                        const float* __restrict__ b2, const int* __restrict__ rowL1,
                        const float* __restrict__ ewL1, const int* __restrict__ map1,
                        const int* __restrict__ scal, float* __restrict__ z) {
    int c = threadIdx.x;
    int m = scal[0];
    float dvm = rsqrtf(deg[m]);
    float acc = b2[c] + dvm * dvm * Q[(long long)map1[m] * HID + c];
    int cnt = imin(scal[1], CAPE1);
    for (int e = 0; e < cnt; ++e) {
        int r = rowL1[e];
        acc += rsqrtf(deg[r]) * ewL1[e] * dvm * Q[(long long)map1[r] * HID + c];
    }
    float mv = ((const float*)scal)[5];
    z[c] = mv * fmaxf(acc, 0.0f);
}

// ---- head MLP: [z|wt|mut|delta|pe](480) -> 512 -> 128 -> 1 ---------------
__global__ void __launch_bounds__(512)
k_head(const float* __restrict__ z, const float* __restrict__ aa,
       const float* __restrict__ pos, const long long* __restrict__ wtI,
       const long long* __restrict__ mutI, const int* __restrict__ scal,
       const float* __restrict__ Wh1, const float* __restrict__ bh1,
       const float* __restrict__ Wh2, const float* __restrict__ bh2,
       const float* __restrict__ Wh3, const float* __restrict__ bh3,
       float* __restrict__ out) {
    __shared__ float feat[480];
    __shared__ float f1[512];
    __shared__ float f2[128];
    int t = threadIdx.x;
    int wi = (int)wtI[0], mi = (int)mutI[0];
    if (t < 256)       feat[t] = z[t];
    else if (t < 320)  feat[t] = aa[wi * 64 + (t - 256)];
    else if (t < 384)  feat[t] = aa[mi * 64 + (t - 320)];
    else if (t < 448)  feat[t] = aa[mi * 64 + (t - 384)] - aa[wi * 64 + (t - 384)];
    else if (t < 480) {
        int m = scal[0];
        int p = m < 0 ? 0 : (m > 511 ? 511 : m);
        feat[t] = pos[p * 32 + (t - 448)];
    }
    __syncthreads();
    float acc = bh1[t];
    for (int j = 0; j < 480; ++j) acc += feat[j] * Wh1[j * 512 + t];
    f1[t] = fmaxf(acc, 0.0f);
    __syncthreads();
    if (t < 128) {
        float a2 = bh2[t];
        for (int j = 0; j < 512; ++j) a2 += f1[j] * Wh2[j * 128 + t];
        f2[t] = fmaxf(a2, 0.0f);
    }
    __syncthreads();
    if (t == 0) {
        float o = bh3[0];
        for (int j = 0; j < 128; ++j) o += f2[j] * Wh3[j];
        out[0] = o;
    }
}

extern "C" void kernel_launch(void* const* d_in, const int* in_sizes, int n_in,
                              void* d_out, int out_size, void* d_ws, size_t ws_size,
                              hipStream_t stream) {
    const float*     x     = (const float*)d_in[0];
    const long long* ei    = (const long long*)d_in[1];
    const float*     ewv   = (const float*)d_in[2];
    const float*     mask  = (const float*)d_in[3];
    const long long* wtI   = (const long long*)d_in[4];
    const long long* mutI  = (const long long*)d_in[5];
    const float*     W1    = (const float*)d_in[6];
    const float*     b1    = (const float*)d_in[7];
    const float*     W2    = (const float*)d_in[8];
    const float*     b2    = (const float*)d_in[9];
    const float*     aa    = (const float*)d_in[10];
    const float*     pos   = (const float*)d_in[11];
    const float*     Wh1   = (const float*)d_in[12];
    const float*     bh1   = (const float*)d_in[13];
    const float*     Wh2   = (const float*)d_in[14];
    const float*     bh2   = (const float*)d_in[15];
    const float*     Wh3   = (const float*)d_in[16];
    const float*     bh3   = (const float*)d_in[17];
    float* out = (float*)d_out;

    const int N = in_sizes[0] / IN_DIM;
    const int E = in_sizes[2];

    // carve workspace (256B aligned slices); total ~5.6 MB
    char* w = (char*)d_ws;
    auto carve = [&](size_t bytes) {
        void* p = (void*)w;
        w += (bytes + 255) & ~(size_t)255;
        return p;
    };
    float* deg   = (float*)carve((size_t)N * 4);
    int*   flag1 = (int*)  carve((size_t)N * 4);
    int*   flag0 = (int*)  carve((size_t)N * 4);
    int*   map1  = (int*)  carve((size_t)N * 4);
    int*   map0  = (int*)  carve((size_t)N * 4);
    int*   scal  = (int*)  carve(64 * 4);
    int*   rowL1 = (int*)  carve(CAPE1 * 4);
    float* ewL1  = (float*)carve(CAPE1 * 4);
    int*   rowL2 = (int*)  carve(CAPE2 * 4);
    int*   colL2 = (int*)  carve(CAPE2 * 4);
    float* ewL2  = (float*)carve(CAPE2 * 4);
    int*   list1 = (int*)  carve(CAP1 * 4);
    int*   list0 = (int*)  carve(CAP0 * 4);
    float* Xg    = (float*)carve((size_t)CAP0 * IN_DIM * 4);
    float* P     = (float*)carve((size_t)CAP0 * HID * 4);
    float* h1    = (float*)carve((size_t)CAP1 * HID * 4);
    float* Q     = (float*)carve((size_t)CAP1 * HID * 4);
    float* z     = (float*)carve(HID * 4);

    const int TB = 256;
    int nb = (N + TB - 1) / TB;
    int eb = (E + TB - 1) / TB;

    k_init    <<<nb, TB, 0, stream>>>(deg, flag1, flag0, scal, N);
    k_findm   <<<nb, TB, 0, stream>>>(mask, flag1, scal, N);
    k_edge1   <<<eb, TB, 0, stream>>>(ei, ewv, deg, flag1, scal, rowL1, ewL1, E);
    k_compact1<<<nb, TB, 0, stream>>>(flag1, flag0, map1, list1, scal, N);
    k_edge2   <<<eb, TB, 0, stream>>>(ei, ewv, flag1, flag0, scal, rowL2, colL2, ewL2, E);
    k_compact0<<<nb, TB, 0, stream>>>(flag0, map0, list0, scal, N);

    k_gatherx<<<CAP0, IN_DIM, 0, stream>>>(x, list0, scal, Xg);
    // P = Xg @ W1  (K=128, out cols=256)
    k_wmma_gemm<<<dim3(CAP0 / 16, HID / 16), 32, 0, stream>>>(Xg, W1, P, IN_DIM, &scal[4], CAP0);

    k_agg1init<<<CAP1, HID, 0, stream>>>(P, deg, b1, list1, map0, scal, h1);
    k_agg1edge<<<CAPE2, HID, 0, stream>>>(P, deg, rowL2, colL2, ewL2, map0, map1, scal, h1);
    k_relu    <<<(CAP1 * HID + TB - 1) / TB, TB, 0, stream>>>(h1, CAP1 * HID);

    // Q = relu(h1) @ W2  (K=256)
    k_wmma_gemm<<<dim3(CAP1 / 16, HID / 16), 32, 0, stream>>>(h1, W2, Q, HID, &scal[3], CAP1);

    k_final   <<<1, HID, 0, stream>>>(Q, deg, b2, rowL1, ewL1, map1, scal, z);
    k_head    <<<1, 512, 0, stream>>>(z, aa, pos, wtI, mutI, scal,
                                      Wh1, bh1, Wh2, bh2, Wh3, bh3, out);
}